// GNN_70970039599600
// MI455X (gfx1250) — compile-verified
//
#include <hip/hip_runtime.h>
#include <hip/hip_bf16.h>

#define HID     128
#define FIN     11
#define FOUT    19
#define NGRAPHS 4096

typedef __attribute__((ext_vector_type(2))) float v2f;
typedef __attribute__((ext_vector_type(8))) float v8f;

// ---------------- degree / norm ----------------
__global__ void k_init_deg(float* deg, int n) {
    int i = blockIdx.x * blockDim.x + threadIdx.x;
    if (i < n) deg[i] = 1.0f;   // self-loop contribution
}

__global__ void k_deg_accum(const int* __restrict__ dst, float* deg, int e) {
    int i = blockIdx.x * blockDim.x + threadIdx.x;
    if (i < e) atomicAdd(&deg[dst[i]], 1.0f);
}

__global__ void k_dinv(const float* __restrict__ deg, float* dinv, int n) {
    int i = blockIdx.x * blockDim.x + threadIdx.x;
    if (i < n) {
        float d = deg[i];
        dinv[i] = (d > 0.0f) ? rsqrtf(d) : 0.0f;
    }
}

// ---------------- layer-1 GEMM: h = x @ W1 (K = 11, scalar) ----------------
__global__ void k_gemm1(const float* __restrict__ x, const float* __restrict__ W1,
                        float* __restrict__ h, int n) {
    int idx = blockIdx.x * blockDim.x + threadIdx.x;
    if (idx >= n * HID) return;
    int node = idx >> 7;
    int f    = idx & (HID - 1);
    const float* xr = x + node * FIN;
    float acc = 0.0f;
#pragma unroll
    for (int k = 0; k < FIN; ++k)
        acc = fmaf(xr[k], W1[k * HID + f], acc);
    h[idx] = acc;
}

// ---------------- out = bias + h * dinv^2 (self-loop message, bias) ----------------
__global__ void k_init_out(float* __restrict__ out, const float* __restrict__ h,
                           const float* __restrict__ bias, const float* __restrict__ dinv,
                           int n) {
    int idx = blockIdx.x * blockDim.x + threadIdx.x;
    if (idx >= n * HID) return;
    int node = idx >> 7;
    int f    = idx & (HID - 1);
    float dn = dinv[node];
    out[idx] = bias[f] + h[idx] * dn * dn;
}

// ---------------- edge scatter: warp-per-edge, float4 gather, fp32 atomics ----------------
__global__ void k_edge_scatter(const float* __restrict__ h, float* __restrict__ out,
                               const int* __restrict__ src, const int* __restrict__ dst,
                               const float* __restrict__ dinv, int e) {
    int gid  = blockIdx.x * blockDim.x + threadIdx.x;
    int edge = gid >> 5;
    int lane = gid & 31;
    if (edge >= e) return;
    int s = src[edge];
    int d = dst[edge];
    float nrm = dinv[s] * dinv[d];
    float4 v = ((const float4*)(h + (size_t)s * HID))[lane];
    float* o = out + (size_t)d * HID + lane * 4;
    atomicAdd(o + 0, v.x * nrm);
    atomicAdd(o + 1, v.y * nrm);
    atomicAdd(o + 2, v.z * nrm);
    atomicAdd(o + 3, v.w * nrm);
}

// ---------------- layer-2 GEMM via V_WMMA_F32_16X16X4_F32 ----------------
// out[n,128] = relu(in[n,128]) @ W[128,128].  One block = 16 rows; 8 waves = 8
// column tiles of 16.  W staged in 64KB LDS.  EXEC stays all-1s around WMMA.
__global__ __launch_bounds__(256) void k_gemm2_wmma(const float* __restrict__ in,
                                                    const float* __restrict__ W,
                                                    float* __restrict__ out, int n) {
    __shared__ float Ws[HID * HID];          // 64 KB
    const int tid = threadIdx.x;

    const float4* W4  = (const float4*)W;
    float4*       Ws4 = (float4*)Ws;
#pragma unroll
    for (int i = 0; i < 16; ++i)             // 4096 float4 / 256 threads
        Ws4[tid + 256 * i] = W4[tid + 256 * i];
    __syncthreads();

    const int m0   = blockIdx.x << 4;
    const int wave = tid >> 5;
    const int lane = tid & 31;
    const int half = lane >> 4;              // 0: K,K+1   1: K+2,K+3
    const int l15  = lane & 15;
    const int n0   = wave << 4;
    int row = m0 + l15;
    if (row >= n) row = n - 1;               // clamp (keeps EXEC full)
    const int ncol = n0 + l15;
    const float* arow = in + (size_t)row * HID;

    v8f c = {};
#pragma unroll
    for (int k0 = 0; k0 < HID; k0 += 4) {
        const int ka = k0 + half * 2;        // even -> 8B aligned
        float2 av = *(const float2*)(arow + ka);
        v2f a, b;
        a[0] = fmaxf(av.x, 0.0f);            // fused relu on layer-1 output
        a[1] = fmaxf(av.y, 0.0f);
        b[0] = Ws[ka * HID + ncol];
        b[1] = Ws[(ka + 1) * HID + ncol];
        c = __builtin_amdgcn_wmma_f32_16x16x4_f32(
                /*neg_a=*/false, a, /*neg_b=*/false, b,
                /*c_mod=*/(short)0, c, /*reuse_a=*/false, /*reuse_b=*/false);
    }

    // D layout: VGPR i -> row m0 + i (+8 for upper lane half), col = ncol
    const int mbase = m0 + half * 8;
#pragma unroll
    for (int i = 0; i < 8; ++i) {
        int r = mbase + i;
        if (r < n) out[(size_t)r * HID + ncol] = c[i];
    }
}

// ---------------- pooling ----------------
__global__ void k_pool_init(float* sums, float* cnt) {
    int idx = blockIdx.x * blockDim.x + threadIdx.x;
    if (idx < NGRAPHS * HID) sums[idx] = 0.0f;
    if (idx < NGRAPHS)       cnt[idx]  = 0.0f;
}

__global__ void k_pool_accum(const float* __restrict__ h, const int* __restrict__ batch,
                             float* sums, int n) {
    int idx = blockIdx.x * blockDim.x + threadIdx.x;
    if (idx >= n * HID) return;
    int node = idx >> 7;
    int f    = idx & (HID - 1);
    atomicAdd(&sums[batch[node] * HID + f], h[idx]);
}

__global__ void k_cnt_accum(const int* __restrict__ batch, float* cnt, int n) {
    int i = blockIdx.x * blockDim.x + threadIdx.x;
    if (i < n) atomicAdd(&cnt[batch[i]], 1.0f);
}

// ---------------- head: (sums/cnt) @ Wlin + blin ----------------
__global__ void k_final(const float* __restrict__ sums, const float* __restrict__ cnt,
                        const float* __restrict__ Wlin, const float* __restrict__ blin,
                        float* __restrict__ out) {
    int idx = blockIdx.x * blockDim.x + threadIdx.x;
    if (idx >= NGRAPHS * FOUT) return;
    int g = idx / FOUT;
    int j = idx - g * FOUT;
    float inv = 1.0f / fmaxf(cnt[g], 1.0f);
    const float* s = sums + g * HID;
    float acc = blin[j];
#pragma unroll 4
    for (int k = 0; k < HID; ++k)
        acc = fmaf(s[k] * inv, Wlin[k * FOUT + j], acc);
    out[idx] = acc;
}

// ---------------- driver ----------------
extern "C" void kernel_launch(void* const* d_in, const int* in_sizes, int n_in,
                              void* d_out, int out_size, void* d_ws, size_t ws_size,
                              hipStream_t stream) {
    (void)n_in; (void)out_size; (void)ws_size;

    const float* x     = (const float*)d_in[0];
    const int*   esrc  = (const int*)  d_in[1];
    const int*   edst  = (const int*)  d_in[2];
    const int*   batch = (const int*)  d_in[3];
    const float* W1    = (const float*)d_in[4];
    const float* b1    = (const float*)d_in[5];
    const float* W2    = (const float*)d_in[6];
    const float* b2    = (const float*)d_in[7];
    const float* Wlin  = (const float*)d_in[8];
    const float* blin  = (const float*)d_in[9];
    float*       out   = (float*)d_out;

    const int N = in_sizes[3];   // batch has N_NODES elements
    const int E = in_sizes[1];

    // carve workspace
    char*  ws  = (char*)d_ws;
    size_t off = 0;
    auto carve = [&](size_t bytes) -> void* {
        off = (off + 255) & ~(size_t)255;
        void* p = ws + off;
        off += bytes;
        return p;
    };
    float* deg   = (float*)carve((size_t)N * 4);
    float* dinv  = (float*)carve((size_t)N * 4);
    float* bufH  = (float*)carve((size_t)N * HID * 4);  // h1, then h2
    float* bufO1 = (float*)carve((size_t)N * HID * 4);  // conv1 output
    float* bufO2 = (float*)carve((size_t)N * HID * 4);  // conv2 output
    float* sums  = (float*)carve((size_t)NGRAPHS * HID * 4);
    float* cnt   = (float*)carve((size_t)NGRAPHS * 4);

    const int T = 256;
    int nb   = (N + T - 1) / T;
    int eb   = (E + T - 1) / T;
    int fb   = (N * HID + T - 1) / T;
    int ewb  = (int)(((size_t)E * 32 + T - 1) / T);   // warp per edge
    int gb2  = (N + 15) / 16;
    int pb   = (NGRAPHS * HID + T - 1) / T;
    int hb   = (NGRAPHS * FOUT + T - 1) / T;

    // degrees & symmetric norm
    k_init_deg <<<nb, T, 0, stream>>>(deg, N);
    k_deg_accum<<<eb, T, 0, stream>>>(edst, deg, E);
    k_dinv     <<<nb, T, 0, stream>>>(deg, dinv, N);

    // layer 1
    k_gemm1       <<<fb,  T, 0, stream>>>(x, W1, bufH, N);
    k_init_out    <<<fb,  T, 0, stream>>>(bufO1, bufH, b1, dinv, N);
    k_edge_scatter<<<ewb, T, 0, stream>>>(bufH, bufO1, esrc, edst, dinv, E);

    // layer 2 (relu fused into WMMA GEMM A-loads)
    k_gemm2_wmma  <<<gb2, T, 0, stream>>>(bufO1, W2, bufH, N);
    k_init_out    <<<fb,  T, 0, stream>>>(bufO2, bufH, b2, dinv, N);
    k_edge_scatter<<<ewb, T, 0, stream>>>(bufH, bufO2, esrc, edst, dinv, E);

    // mean pool + head
    k_pool_init <<<pb, T, 0, stream>>>(sums, cnt);
    k_pool_accum<<<fb, T, 0, stream>>>(bufO2, batch, sums, N);
    k_cnt_accum <<<nb, T, 0, stream>>>(batch, cnt, N);
    k_final     <<<hb, T, 0, stream>>>(sums, cnt, Wlin, blin, out);
}